// ModelNew_4647154615203
// MI455X (gfx1250) — compile-verified
//
#include <hip/hip_runtime.h>

typedef __attribute__((ext_vector_type(2))) float v2f;
typedef __attribute__((ext_vector_type(8))) float v8f;

#define S_LEN 2048
#define HID   1024
#define NH    16
#define NKV   4
#define HD    64

// ---------------------------------------------------------------------------
// GEMM: C[M,N] = A[M,K] @ B[K,N], fp32, row-major, via V_WMMA_F32_16X16X4_F32.
// Block = 256 threads (8 waves). Wave tile = 16(M) x 64(N): 4 accumulators,
// A fragment reused across the 4 N-tiles. K stepped by 4 (WMMA depth).
// A frag (16x4):  lane L -> m = L%16; comps = K { k0, k0+1 | k0+2, k0+3 } by half.
// B frag (4x16):  lane L -> n = L%16; comps = same K split by half.
// C/D (16x16):    VGPR r, lanes 0-15 = row r, lanes 16-31 = row r+8.
// ---------------------------------------------------------------------------
__global__ __launch_bounds__(256) void gemm_wmma_f32(
    const float* __restrict__ A, const float* __restrict__ B,
    float* __restrict__ C, int M, int N, int K)
{
    const int lane = threadIdx.x & 31;
    const int wave = threadIdx.x >> 5;
    const int half = lane >> 4;
    const int l16  = lane & 15;

    const int tileM = blockIdx.y * 128 + wave * 16;
    const int tileN = blockIdx.x * 64;

    v8f acc0 = {}, acc1 = {}, acc2 = {}, acc3 = {};
    const float* arow = A + (size_t)(tileM + l16) * K + 2 * half;

    #pragma unroll 4
    for (int k0 = 0; k0 < K; k0 += 4) {
        v2f a = *(const v2f*)(arow + k0);
        const float* brow = B + (size_t)(k0 + 2 * half) * N + tileN + l16;
        v2f b0, b1, b2, b3;
        b0.x = brow[0];  b0.y = brow[N];
        b1.x = brow[16]; b1.y = brow[N + 16];
        b2.x = brow[32]; b2.y = brow[N + 32];
        b3.x = brow[48]; b3.y = brow[N + 48];
        acc0 = __builtin_amdgcn_wmma_f32_16x16x4_f32(false, a, false, b0, (short)0, acc0, false, false);
        acc1 = __builtin_amdgcn_wmma_f32_16x16x4_f32(false, a, false, b1, (short)0, acc1, false, false);
        acc2 = __builtin_amdgcn_wmma_f32_16x16x4_f32(false, a, false, b2, (short)0, acc2, false, false);
        acc3 = __builtin_amdgcn_wmma_f32_16x16x4_f32(false, a, false, b3, (short)0, acc3, false, false);
    }

    float* crow = C + (size_t)(tileM + 8 * half) * N + tileN + l16;
    #pragma unroll
    for (int r = 0; r < 8; ++r) {
        crow[(size_t)r * N +  0] = acc0[r];
        crow[(size_t)r * N + 16] = acc1[r];
        crow[(size_t)r * N + 32] = acc2[r];
        crow[(size_t)r * N + 48] = acc3[r];
    }
}

// ---------------------------------------------------------------------------
// RoPE (in place), matching the reference's table: pair i uses
// inv_freq[(2i) mod 32], inv_freq[j] = 10000^(-j/32).
// x layout: [S, nheads*64]; grid.x = S; block = nheads*32 (one thread / pair).
// ---------------------------------------------------------------------------
__global__ void rope_f32(float* __restrict__ x, int nheads)
{
    const int s = blockIdx.x;
    const int h = threadIdx.x >> 5;
    const int i = threadIdx.x & 31;

    const float fidx = (float)((2 * i) & 31);
    const float inv_freq = powf(10000.0f, -fidx * (1.0f / 32.0f));
    const float ang = (float)s * inv_freq;
    float sn, cs;
    sincosf(ang, &sn, &cs);

    float* p = x + (size_t)s * (size_t)(nheads * HD) + h * HD + 2 * i;
    const float xe = p[0], xo = p[1];
    p[0] = xe * cs - xo * sn;
    p[1] = xe * sn + xo * cs;
}

// ---------------------------------------------------------------------------
// Causal GQA attention, flash-style, fp32 WMMA.
// Grid: (S/128, NH). Block: 256 threads = 8 waves; wave w owns query rows
// [q0, q0+16). Q kept in registers as 16 A-fragments. Per 16-key block:
//   S = Q K^T (16 WMMAs) -> scale + causal mask -> online softmax
//   (row reductions via shfl_xor within the 16-lane half that owns the row)
//   P staged through per-wave LDS tile to re-layout C->A, then O += P V
//   (16 WMMAs). No cross-wave sync (per-wave causal trip counts differ).
// ---------------------------------------------------------------------------
__global__ __launch_bounds__(256) void attn_wmma_f32(
    const float* __restrict__ Q, const float* __restrict__ Kp,
    const float* __restrict__ V, float* __restrict__ O)
{
    __shared__ float p_lds[8][16][17];  // per-wave 16x16 P tile, padded

    const int lane = threadIdx.x & 31;
    const int wave = threadIdx.x >> 5;
    const int half = lane >> 4;
    const int l16  = lane & 15;

    const int h  = blockIdx.y;
    const int kv = h >> 2;                       // G = NH/NKV = 4
    const int q0 = blockIdx.x * 128 + wave * 16;

    // Q tile (16 rows x 64 dims) as 16 K-step fragments
    v2f qa[16];
    const float* qb = Q + (size_t)(q0 + l16) * (NH * HD) + h * HD + 2 * half;
    #pragma unroll
    for (int kk = 0; kk < 16; ++kk)
        qa[kk] = *(const v2f*)(qb + 4 * kk);

    v8f o0 = {}, o1 = {}, o2 = {}, o3 = {};
    float m_run[8], l_run[8];
    #pragma unroll
    for (int r = 0; r < 8; ++r) { m_run[r] = -3.0e38f; l_run[r] = 0.0f; }

    for (int j0 = 0; j0 <= q0; j0 += 16) {
        // ---- scores: S(16x16) = Q * K^T -------------------------------
        const float* kb = Kp + (size_t)(j0 + l16) * (NKV * HD) + kv * HD + 2 * half;
        v8f sacc = {};
        #pragma unroll
        for (int kk = 0; kk < 16; ++kk) {
            v2f b = *(const v2f*)(kb + 4 * kk);
            sacc = __builtin_amdgcn_wmma_f32_16x16x4_f32(false, qa[kk], false, b, (short)0, sacc, false, false);
        }

        // ---- scale, causal mask, online softmax -----------------------
        #pragma unroll
        for (int r = 0; r < 8; ++r) {
            const int qrow = q0 + r + 8 * half;
            const int kcol = j0 + l16;
            float sc = sacc[r] * 0.125f;                 // 1/sqrt(64)
            sc = (kcol > qrow) ? -1.0e30f : sc;

            float rmax = sc;
            rmax = fmaxf(rmax, __shfl_xor(rmax, 8, 32));
            rmax = fmaxf(rmax, __shfl_xor(rmax, 4, 32));
            rmax = fmaxf(rmax, __shfl_xor(rmax, 2, 32));
            rmax = fmaxf(rmax, __shfl_xor(rmax, 1, 32));

            const float mnew  = fmaxf(m_run[r], rmax);
            const float alpha = expf(m_run[r] - mnew);
            const float pv    = expf(sc - mnew);

            float rsum = pv;
            rsum += __shfl_xor(rsum, 8, 32);
            rsum += __shfl_xor(rsum, 4, 32);
            rsum += __shfl_xor(rsum, 2, 32);
            rsum += __shfl_xor(rsum, 1, 32);

            l_run[r] = l_run[r] * alpha + rsum;
            m_run[r] = mnew;
            o0[r] *= alpha; o1[r] *= alpha; o2[r] *= alpha; o3[r] *= alpha;

            p_lds[wave][r + 8 * half][l16] = pv;         // C-layout -> LDS
        }
        asm volatile("" ::: "memory");  // DS ops are in-order per wave; block compiler reorder

        // ---- O(16x64) += P(16x16) * V(16x64) --------------------------
        #pragma unroll
        for (int kk = 0; kk < 4; ++kk) {
            v2f pa;
            pa.x = p_lds[wave][l16][4 * kk + 2 * half];
            pa.y = p_lds[wave][l16][4 * kk + 2 * half + 1];
            const int krow = j0 + 4 * kk + 2 * half;
            const float* vb = V + (size_t)krow * (NKV * HD) + kv * HD + l16;
            v2f b0, b1, b2, b3;
            b0.x = vb[0];  b0.y = vb[NKV * HD];
            b1.x = vb[16]; b1.y = vb[NKV * HD + 16];
            b2.x = vb[32]; b2.y = vb[NKV * HD + 32];
            b3.x = vb[48]; b3.y = vb[NKV * HD + 48];
            o0 = __builtin_amdgcn_wmma_f32_16x16x4_f32(false, pa, false, b0, (short)0, o0, false, false);
            o1 = __builtin_amdgcn_wmma_f32_16x16x4_f32(false, pa, false, b1, (short)0, o1, false, false);
            o2 = __builtin_amdgcn_wmma_f32_16x16x4_f32(false, pa, false, b2, (short)0, o2, false, false);
            o3 = __builtin_amdgcn_wmma_f32_16x16x4_f32(false, pa, false, b3, (short)0, o3, false, false);
        }
        asm volatile("" ::: "memory");
    }

    // ---- normalize and store [s, h*64+d] ------------------------------
    float* ob = O + (size_t)(q0 + 8 * half) * (NH * HD) + h * HD + l16;
    #pragma unroll
    for (int r = 0; r < 8; ++r) {
        const float inv_l = 1.0f / l_run[r];
        ob[(size_t)r * (NH * HD) +  0] = o0[r] * inv_l;
        ob[(size_t)r * (NH * HD) + 16] = o1[r] * inv_l;
        ob[(size_t)r * (NH * HD) + 32] = o2[r] * inv_l;
        ob[(size_t)r * (NH * HD) + 48] = o3[r] * inv_l;
    }
}

// ---------------------------------------------------------------------------
extern "C" void kernel_launch(void* const* d_in, const int* in_sizes, int n_in,
                              void* d_out, int out_size, void* d_ws, size_t ws_size,
                              hipStream_t stream)
{
    const float* X  = (const float*)d_in[0];   // [1, 2048, 1024]
    const float* Wq = (const float*)d_in[1];   // [1024, 1024]
    const float* Wk = (const float*)d_in[2];   // [1024, 256]
    const float* Wv = (const float*)d_in[3];   // [1024, 256]
    const float* Wo = (const float*)d_in[4];   // [1024, 1024]
    float* out = (float*)d_out;                // [1, 2048, 1024]

    float* q    = (float*)d_ws;                         // 2048*1024
    float* k    = q + (size_t)S_LEN * NH  * HD;         // 2048*256
    float* v    = k + (size_t)S_LEN * NKV * HD;         // 2048*256
    float* attn = v + (size_t)S_LEN * NKV * HD;         // 2048*1024  (20 MB total)

    dim3 blk(256);
    gemm_wmma_f32<<<dim3((NH  * HD) / 64, S_LEN / 128), blk, 0, stream>>>(X, Wq, q, S_LEN, NH  * HD, HID);
    gemm_wmma_f32<<<dim3((NKV * HD) / 64, S_LEN / 128), blk, 0, stream>>>(X, Wk, k, S_LEN, NKV * HD, HID);
    gemm_wmma_f32<<<dim3((NKV * HD) / 64, S_LEN / 128), blk, 0, stream>>>(X, Wv, v, S_LEN, NKV * HD, HID);

    rope_f32<<<S_LEN, NH  * 32, 0, stream>>>(q, NH);
    rope_f32<<<S_LEN, NKV * 32, 0, stream>>>(k, NKV);

    attn_wmma_f32<<<dim3(S_LEN / 128, NH), blk, 0, stream>>>(q, k, v, attn);

    gemm_wmma_f32<<<dim3(HID / 64, S_LEN / 128), blk, 0, stream>>>(attn, Wo, out, S_LEN, HID, HID);
}